// MaskedAveragePooling_48438641164895
// MI455X (gfx1250) — compile-verified
//
#include <hip/hip_runtime.h>

#define NUM_INST 64
#define CH       256
#define OUTD     64

// workspace layout (floats)
#define WS_SUMF  0                       // [64][256] feature sums
#define WS_CNT   (NUM_INST * CH)         // [64] counts
#define WS_SUMC  (WS_CNT + NUM_INST)     // [64][3] coord sums
#define WS_TOTAL (WS_SUMC + NUM_INST * 3)

typedef float v2f __attribute__((ext_vector_type(2)));
typedef float v8f __attribute__((ext_vector_type(8)));

// ---------------------------------------------------------------- zero ws
__global__ void zero_ws_kernel(float* __restrict__ ws) {
    int i = blockIdx.x * blockDim.x + threadIdx.x;
    if (i < WS_TOTAL) ws[i] = 0.0f;
}

// ------------------------------------------------- segment-sum reduction
// one wave per point: 2x float4 per lane = 256 coalesced channels,
// LDS-privatized accumulators, global f32 atomic flush per block.
__global__ __launch_bounds__(256) void seg_reduce_kernel(
    const float* __restrict__ features, const float* __restrict__ coords,
    const int* __restrict__ ids, float* __restrict__ ws, int n) {
    __shared__ float lsum[NUM_INST * CH];   // 64 KB
    __shared__ float lcnt[NUM_INST];
    __shared__ float lco[NUM_INST * 3];

    for (int t = threadIdx.x; t < NUM_INST * CH; t += blockDim.x) lsum[t] = 0.0f;
    for (int t = threadIdx.x; t < NUM_INST;      t += blockDim.x) lcnt[t] = 0.0f;
    for (int t = threadIdx.x; t < NUM_INST * 3;  t += blockDim.x) lco[t]  = 0.0f;
    __syncthreads();

    const int lane = threadIdx.x & 31;
    const int wid  = threadIdx.x >> 5;
    const int wpb  = blockDim.x >> 5;
    const int gw   = blockIdx.x * wpb + wid;
    const int gstep = gridDim.x * wpb;

    for (int p = gw; p < n; p += gstep) {
        const int id = ids[p];
        if (id < 0) continue;                       // background
        const float4* row = (const float4*)(features + (size_t)p * CH);
        float4 f0 = row[lane];                      // channels lane*4 .. +3
        float4 f1 = row[lane + 32];                 // channels 128+lane*4 ..
        float* dst = lsum + id * CH;
        const int c0 = lane * 4;
        atomicAdd(dst + c0 + 0, f0.x);
        atomicAdd(dst + c0 + 1, f0.y);
        atomicAdd(dst + c0 + 2, f0.z);
        atomicAdd(dst + c0 + 3, f0.w);
        atomicAdd(dst + 128 + c0 + 0, f1.x);
        atomicAdd(dst + 128 + c0 + 1, f1.y);
        atomicAdd(dst + 128 + c0 + 2, f1.z);
        atomicAdd(dst + 128 + c0 + 3, f1.w);
        if (lane == 0) {
            atomicAdd(&lcnt[id], 1.0f);
        } else if (lane < 4) {
            atomicAdd(&lco[id * 3 + (lane - 1)], coords[(size_t)p * 3 + (lane - 1)]);
        }
    }
    __syncthreads();

    for (int t = threadIdx.x; t < NUM_INST * CH; t += blockDim.x)
        atomicAdd(ws + WS_SUMF + t, lsum[t]);
    for (int t = threadIdx.x; t < NUM_INST;      t += blockDim.x)
        atomicAdd(ws + WS_CNT + t, lcnt[t]);
    for (int t = threadIdx.x; t < NUM_INST * 3;  t += blockDim.x)
        atomicAdd(ws + WS_SUMC + t, lco[t]);
}

// ------------------------------------------------------------- finalize
// embeddings = sumF / max(cnt,1) -> d_out[0:16384]
// centroids  = sumC / max(cnt,1) -> d_out[16384:16576]
__global__ void finalize_kernel(const float* __restrict__ ws, float* __restrict__ out) {
    const int inst = blockIdx.x;
    const float inv = 1.0f / fmaxf(ws[WS_CNT + inst], 1.0f);
    for (int t = threadIdx.x; t < CH; t += blockDim.x)
        out[inst * CH + t] = ws[WS_SUMF + inst * CH + t] * inv;
    if (threadIdx.x < 3)
        out[NUM_INST * CH + inst * 3 + threadIdx.x] =
            ws[WS_SUMC + inst * 3 + threadIdx.x] * inv;
}

// ------------------------------------------------------- MLP via f32 WMMA
// 16 waves, each owns one 16x16 tile of the 64x64 result of each layer.
// V_WMMA_F32_16X16X4_F32 lane layout:
//   A (16x4): lanes 0-15 -> M=lane, K=k..k+1 ; lanes 16-31 -> M=lane-16, K=k+2..k+3
//   B (4x16): lanes 0-15 -> N=lane, K=k..k+1 ; lanes 16-31 -> N=lane-16, K=k+2..k+3
//   C/D     : vgpr r -> M = 16*mi + r + 8*(lane>=16), N = 16*ni + (lane&15)
__global__ __launch_bounds__(512) void mlp_wmma_kernel(
    const float* __restrict__ emb,   // [64,256]  (d_out base)
    const float* __restrict__ W1,    // [256,64]
    const float* __restrict__ W2,    // [64,64]
    const float* __restrict__ W3,    // [64,64]
    const float* __restrict__ b3,    // [64]
    float* __restrict__ out) {       // [64,64]
    __shared__ float h1[64 * 64];
    __shared__ float h2[64 * 64];

    const int lane  = threadIdx.x & 31;
    const int wid   = threadIdx.x >> 5;      // 0..15
    const int mbase = (wid >> 2) * 16;
    const int nbase = (wid & 3) * 16;
    const int half  = lane >> 4;
    const int l15   = lane & 15;
    const int am    = mbase + l15;           // A row this lane supplies
    const int bn    = nbase + l15;           // B col this lane supplies

    // ---- layer 1: relu(emb @ W1) -> h1
    v8f c1 = {};
    for (int k = 0; k < CH; k += 4) {
        const int ka = k + 2 * half;
        v2f a, b;
        a.x = emb[am * CH + ka];
        a.y = emb[am * CH + ka + 1];
        b.x = W1[ka * 64 + bn];
        b.y = W1[(ka + 1) * 64 + bn];
        c1 = __builtin_amdgcn_wmma_f32_16x16x4_f32(false, a, false, b,
                                                   (short)0, c1, false, false);
    }
#pragma unroll
    for (int r = 0; r < 8; ++r)
        h1[(mbase + r + 8 * half) * 64 + bn] = fmaxf(c1[r], 0.0f);
    __syncthreads();

    // ---- layer 2: relu(h1 @ W2) -> h2
    v8f c2 = {};
    for (int k = 0; k < 64; k += 4) {
        const int ka = k + 2 * half;
        v2f a, b;
        a.x = h1[am * 64 + ka];
        a.y = h1[am * 64 + ka + 1];
        b.x = W2[ka * 64 + bn];
        b.y = W2[(ka + 1) * 64 + bn];
        c2 = __builtin_amdgcn_wmma_f32_16x16x4_f32(false, a, false, b,
                                                   (short)0, c2, false, false);
    }
#pragma unroll
    for (int r = 0; r < 8; ++r)
        h2[(mbase + r + 8 * half) * 64 + bn] = fmaxf(c2[r], 0.0f);
    __syncthreads();

    // ---- layer 3: h2 @ W3 + b3 -> out
    v8f c3 = {};
    for (int k = 0; k < 64; k += 4) {
        const int ka = k + 2 * half;
        v2f a, b;
        a.x = h2[am * 64 + ka];
        a.y = h2[am * 64 + ka + 1];
        b.x = W3[ka * 64 + bn];
        b.y = W3[(ka + 1) * 64 + bn];
        c3 = __builtin_amdgcn_wmma_f32_16x16x4_f32(false, a, false, b,
                                                   (short)0, c3, false, false);
    }
    const float bias = b3[bn];
#pragma unroll
    for (int r = 0; r < 8; ++r)
        out[(mbase + r + 8 * half) * 64 + bn] = c3[r] + bias;
}

// ----------------------------------------------------------------- launch
extern "C" void kernel_launch(void* const* d_in, const int* in_sizes, int n_in,
                              void* d_out, int out_size, void* d_ws, size_t ws_size,
                              hipStream_t stream) {
    (void)n_in; (void)out_size; (void)ws_size;
    const float* features = (const float*)d_in[0];
    const float* coords   = (const float*)d_in[1];
    const int*   ids      = (const int*)d_in[2];
    const float* W1       = (const float*)d_in[3];
    const float* W2       = (const float*)d_in[4];
    const float* W3       = (const float*)d_in[5];
    const float* b3       = (const float*)d_in[6];
    float* out = (float*)d_out;
    float* ws  = (float*)d_ws;
    const int n = in_sizes[2];   // number of points (instance_ids count)

    zero_ws_kernel<<<(WS_TOTAL + 255) / 256, 256, 0, stream>>>(ws);
    seg_reduce_kernel<<<2048, 256, 0, stream>>>(features, coords, ids, ws, n);
    finalize_kernel<<<NUM_INST, 256, 0, stream>>>(ws, out);
    mlp_wmma_kernel<<<1, 512, 0, stream>>>(out, W1, W2, W3, b3,
                                           out + NUM_INST * CH + NUM_INST * 3);
}